// PoM_22986664968549
// MI455X (gfx1250) — compile-verified
//
#include <hip/hip_runtime.h>

typedef float v2f __attribute__((ext_vector_type(2)));
typedef float v8f __attribute__((ext_vector_type(8)));

#define BB 4
#define NN 4096
#define MM (BB * NN)   // 16384 rows total
#define DIMC 128
#define DEC 512
#define HALF_DE 256

// ---------------------------------------------------------------------------
// fp32 WMMA helper: D = A(16x4) * B(4x16) + C, full fp32 (matches reference).
// ---------------------------------------------------------------------------
static __device__ __forceinline__ v8f wmma_f32(v2f a, v2f b, v8f c) {
  return __builtin_amdgcn_wmma_f32_16x16x4_f32(
      /*neg_a=*/false, a, /*neg_b=*/false, b,
      /*c_mod=*/(short)0, c, /*reuse_a=*/false, /*reuse_b=*/false);
}

static __device__ __forceinline__ float gelu_erf(float x) {
  return 0.5f * x * (1.0f + erff(x * 0.70710678118654752440f));
}
static __device__ __forceinline__ float sigmoidf(float x) {
  return 1.0f / (1.0f + __expf(-x));
}

// ---------------------------------------------------------------------------
// Kernel 0: transpose W_ag [128,512] -> Wt [512,128] so k3 B-frags coalesce.
// ---------------------------------------------------------------------------
__global__ __launch_bounds__(128) void k0_transpose_wag(
    const float* __restrict__ W_ag, float* __restrict__ Wt) {
  int idx = blockIdx.x * 128 + threadIdx.x;  // 0 .. 128*512-1
  int j = idx >> 9;                          // row of W_ag (0..127)
  int d = idx & 511;                         // col of W_ag (0..511)
  Wt[(size_t)d * DIMC + j] = W_ag[idx];
}

// ---------------------------------------------------------------------------
// Kernel 1: fused dual projection  s = xq@W_se^T+b_se,  h = xq@W_po^T+b_po
// then epilogue: sig = sigmoid(s);  hp = [g1, g1*g2] with g = gelu_erf(h).
// Block: 128 thr = 4 waves, covers 64 M-rows x column-pair (d0, d0+256).
// fp32 WMMA 16x16x4, K staged in 64-wide LDS chunks.
// ---------------------------------------------------------------------------
#define K1_ST 68  // 64 + 4 pad: conflict-free ds_load_b64, 16B-aligned rows

__global__ __launch_bounds__(128) void k1_proj_act(
    const float* __restrict__ xq,
    const float* __restrict__ W_se, const float* __restrict__ b_se,
    const float* __restrict__ W_po, const float* __restrict__ b_po,
    float* __restrict__ sig, float* __restrict__ hp) {
  __shared__ float lA[64 * K1_ST];
  __shared__ float lW[4][16 * K1_ST];

  const int m0 = blockIdx.x * 64;
  const int d0 = blockIdx.y * 16;   // 0..240
  const int d1 = d0 + HALF_DE;      // paired half

  const int t     = threadIdx.x;
  const int w     = t >> 5;
  const int lane  = t & 31;
  const int row   = lane & 15;
  const int n     = row;
  const int khalf = (lane >> 4) << 1;  // 0 or 2 (A/B K split across halves)

  const float* w0 = W_se + (size_t)d0 * DIMC;
  const float* w1 = W_se + (size_t)d1 * DIMC;
  const float* w2 = W_po + (size_t)d0 * DIMC;
  const float* w3 = W_po + (size_t)d1 * DIMC;

  v8f acc_s0 = {}; v8f acc_s1 = {}; v8f acc_h0 = {}; v8f acc_h1 = {};

  for (int kc = 0; kc < DIMC; kc += 64) {
    // stage A chunk: 64 rows x 64 K of xq (row-major, 256B/row contiguous)
    #pragma unroll
    for (int i = 0; i < 8; ++i) {
      int fi = (i * 128 + t) * 4; int r = fi >> 6; int c = fi & 63;
      *(float4*)&lA[r * K1_ST + c] =
          *(const float4*)&xq[(size_t)(m0 + r) * DIMC + kc + c];
    }
    // stage 4 weight tiles: 16 rows x 64 K each
    {
      const float* srcs[4] = {w0, w1, w2, w3};
      #pragma unroll
      for (int mtx = 0; mtx < 4; ++mtx) {
        #pragma unroll
        for (int i = 0; i < 2; ++i) {
          int fi = (i * 128 + t) * 4; int r = fi >> 6; int c = fi & 63;
          *(float4*)&lW[mtx][r * K1_ST + c] =
              *(const float4*)&srcs[mtx][(size_t)r * DIMC + kc + c];
        }
      }
    }
    if (kc + 64 < DIMC)
      __builtin_prefetch(&xq[(size_t)m0 * DIMC + kc + 64], 0, 0);
    __syncthreads();

    #pragma unroll
    for (int k0 = 0; k0 < 64; k0 += 4) {
      int kb = k0 + khalf;
      v2f a  = *(const v2f*)&lA[(w * 16 + row) * K1_ST + kb];
      v2f b0 = *(const v2f*)&lW[0][n * K1_ST + kb];
      v2f b1 = *(const v2f*)&lW[1][n * K1_ST + kb];
      v2f b2 = *(const v2f*)&lW[2][n * K1_ST + kb];
      v2f b3 = *(const v2f*)&lW[3][n * K1_ST + kb];
      acc_s0 = wmma_f32(a, b0, acc_s0);
      acc_s1 = wmma_f32(a, b1, acc_s1);
      acc_h0 = wmma_f32(a, b2, acc_h0);
      acc_h1 = wmma_f32(a, b3, acc_h1);
    }
    __syncthreads();
  }

  // epilogue: bias + sigmoid / gelu + polynomial pairing
  const int hi8 = (lane >> 4) * 8;
  const float bs0 = b_se[d0 + n], bs1 = b_se[d1 + n];
  const float bp0 = b_po[d0 + n], bp1 = b_po[d1 + n];
  #pragma unroll
  for (int r = 0; r < 8; ++r) {
    int m = m0 + w * 16 + r + hi8;
    float g1 = gelu_erf(acc_h0[r] + bp0);
    float g2 = gelu_erf(acc_h1[r] + bp1);
    size_t o0 = (size_t)m * DEC + d0 + n;
    size_t o1 = (size_t)m * DEC + d1 + n;
    sig[o0] = sigmoidf(acc_s0[r] + bs0);
    sig[o1] = sigmoidf(acc_s1[r] + bs1);
    hp[o0]  = g1;
    hp[o1]  = g1 * g2;
  }
}

// ---------------------------------------------------------------------------
// Kernel 2: causal running-mean (the masked mixer, exploiting tril structure)
// as a 3-phase chunked scan over N (64 chunks of 64 rows per batch).
// ---------------------------------------------------------------------------
__global__ __launch_bounds__(128) void k2a_chunk_sums(
    const float* __restrict__ hp, float* __restrict__ P) {
  int c = blockIdx.x, g = blockIdx.y, b = blockIdx.z;
  int d = g * 128 + threadIdx.x;
  const float* src = hp + ((size_t)(b * NN + c * 64)) * DEC + d;
  float s = 0.f;
  #pragma unroll 4
  for (int r = 0; r < 64; ++r) s += src[(size_t)r * DEC];
  P[((size_t)(b * 64 + c)) * DEC + d] = s;
}

__global__ __launch_bounds__(128) void k2b_scan_partials(float* __restrict__ P) {
  int g = blockIdx.x, b = blockIdx.y;
  int d = g * 128 + threadIdx.x;
  float run = 0.f;
  for (int c = 0; c < 64; ++c) {
    size_t idx = ((size_t)(b * 64 + c)) * DEC + d;
    float v = P[idx];
    P[idx] = run;     // exclusive prefix
    run += v;
  }
}

__global__ __launch_bounds__(128) void k2c_gate(
    float* __restrict__ hp, const float* __restrict__ sig,
    const float* __restrict__ P) {
  int c = blockIdx.x, g = blockIdx.y, b = blockIdx.z;
  int d = g * 128 + threadIdx.x;
  float acc = P[((size_t)(b * 64 + c)) * DEC + d];
  size_t base = ((size_t)(b * NN + c * 64)) * DEC + d;
  for (int r = 0; r < 64; ++r) {
    size_t idx = base + (size_t)r * DEC;
    acc += hp[idx];
    int nl = c * 64 + r;  // row within batch
    float inv = 1.0f / (1e-7f + (float)(nl + 1));
    hp[idx] = sig[idx] * acc * inv;  // overwrite hp with gated activations
  }
}

// ---------------------------------------------------------------------------
// Kernel 3: out = gated @ W_ag^T + b_ag   ([16384,512] x [512,128], fp32 WMMA)
// Block: 128 thr = 4 waves, one 16-row M tile x all 128 output cols.
// ---------------------------------------------------------------------------
#define K3_AS 68   // 64 + 4 pad
#define K3_BS 132  // 128 + 4 pad (keeps float4 LDS stores 16B aligned)

__global__ __launch_bounds__(128) void k3_out_gemm(
    const float* __restrict__ gated, const float* __restrict__ Wt,
    const float* __restrict__ b_ag, float* __restrict__ out) {
  __shared__ float lA[16 * K3_AS];
  __shared__ float lB[64 * K3_BS];

  const int m0    = blockIdx.x * 16;
  const int t     = threadIdx.x;
  const int w     = t >> 5;
  const int lane  = t & 31;
  const int row   = lane & 15;
  const int n     = row;
  const int khalf = (lane >> 4) << 1;
  const int jA = w * 16, jB = jA + 64;

  v8f accA = {}; v8f accB = {};

  for (int kc = 0; kc < DEC; kc += 64) {
    // stage A: 16 rows x 64 K of gated
    #pragma unroll
    for (int i = 0; i < 2; ++i) {
      int fi = (i * 128 + t) * 4; int r = fi >> 6; int c = fi & 63;
      *(float4*)&lA[r * K3_AS + c] =
          *(const float4*)&gated[(size_t)(m0 + r) * DEC + kc + c];
    }
    // stage B: Wt rows kc..kc+63 x 128 cols (fully contiguous region)
    #pragma unroll
    for (int i = 0; i < 16; ++i) {
      int fi = (i * 128 + t) * 4; int r = fi >> 7; int c = fi & 127;
      *(float4*)&lB[r * K3_BS + c] =
          *(const float4*)&Wt[(size_t)(kc + r) * DIMC + c];
    }
    if (kc + 64 < DEC)
      __builtin_prefetch(&Wt[(size_t)(kc + 64) * DIMC], 0, 0);
    __syncthreads();

    #pragma unroll
    for (int k0 = 0; k0 < 64; k0 += 4) {
      int kb = k0 + khalf;
      v2f a  = *(const v2f*)&lA[row * K3_AS + kb];
      v2f b0 = {lB[kb * K3_BS + jA + n], lB[(kb + 1) * K3_BS + jA + n]};
      v2f b1 = {lB[kb * K3_BS + jB + n], lB[(kb + 1) * K3_BS + jB + n]};
      accA = wmma_f32(a, b0, accA);
      accB = wmma_f32(a, b1, accB);
    }
    __syncthreads();
  }

  const int hi8 = (lane >> 4) * 8;
  const float biasA = b_ag[jA + n], biasB = b_ag[jB + n];
  #pragma unroll
  for (int r = 0; r < 8; ++r) {
    int m = m0 + r + hi8;
    out[(size_t)m * DIMC + jA + n] = accA[r] + biasA;
    out[(size_t)m * DIMC + jB + n] = accB[r] + biasB;
  }
}

// ---------------------------------------------------------------------------
// Launch. Inputs: xq, mask(unused: causal structure exploited analytically),
// W_se, b_se, W_po, b_po, W_ag, b_ag.  Output fp32 [B,N,128].
// Workspace: sig(32MB) | hp/gated(32MB) | partials(512KB) | Wt(256KB).
// ---------------------------------------------------------------------------
extern "C" void kernel_launch(void* const* d_in, const int* in_sizes, int n_in,
                              void* d_out, int out_size, void* d_ws,
                              size_t ws_size, hipStream_t stream) {
  (void)in_sizes; (void)n_in; (void)out_size; (void)ws_size;
  const float* xq   = (const float*)d_in[0];
  const float* W_se = (const float*)d_in[2];
  const float* b_se = (const float*)d_in[3];
  const float* W_po = (const float*)d_in[4];
  const float* b_po = (const float*)d_in[5];
  const float* W_ag = (const float*)d_in[6];
  const float* b_ag = (const float*)d_in[7];
  float* out = (float*)d_out;

  float* ws  = (float*)d_ws;
  float* sig = ws;                               // MM*DE
  float* hp  = sig + (size_t)MM * DEC;           // MM*DE (becomes gated)
  float* P   = hp + (size_t)MM * DEC;            // BB*64*DE partials
  float* Wt  = P + (size_t)BB * 64 * DEC;        // DE*DIM transposed W_ag

  k0_transpose_wag<<<dim3(DIMC * DEC / 128), dim3(128), 0, stream>>>(W_ag, Wt);
  k1_proj_act<<<dim3(MM / 64, 16), dim3(128), 0, stream>>>(
      xq, W_se, b_se, W_po, b_po, sig, hp);
  k2a_chunk_sums<<<dim3(64, 4, BB), dim3(128), 0, stream>>>(hp, P);
  k2b_scan_partials<<<dim3(4, BB), dim3(128), 0, stream>>>(P);
  k2c_gate<<<dim3(64, 4, BB), dim3(128), 0, stream>>>(hp, sig, P);
  k3_out_gemm<<<dim3(MM / 16), dim3(128), 0, stream>>>(hp, Wt, b_ag, out);
}